// Model_72902774882652
// MI455X (gfx1250) — compile-verified
//
#include <hip/hip_runtime.h>

// ---------------------------------------------------------------------------
// CDNA5 (gfx1250) async global->LDS helpers.  Prefer clang builtins when the
// toolchain declares them; otherwise fall back to inline CDNA5 assembly.
// The builtin signature (from hipcc diagnostics) is:
//   void __builtin_amdgcn_global_load_async_to_lds_b128(
//       int4 AS(1)* src, int4 AS(3)* dst, Imm int offset, Imm int cpol)
// ---------------------------------------------------------------------------
#define ASG __attribute__((address_space(1)))
#define ASL __attribute__((address_space(3)))

typedef int v4i_cd5 __attribute__((ext_vector_type(4)));

#if __has_builtin(__builtin_amdgcn_global_load_async_to_lds_b128)
#define ASYNC_COPY_B128(gsrc, ldst)                                            \
  do {                                                                         \
    __builtin_amdgcn_global_load_async_to_lds_b128(                            \
        (ASG v4i_cd5*)(gsrc), (ASL v4i_cd5*)(ldst), 0, 0);                     \
  } while (0)
#else
#define ASYNC_COPY_B128(gsrc, ldst)                                            \
  asm volatile("global_load_async_to_lds_b128 %0, %1, off"                     \
               :: "v"((unsigned)(unsigned long long)(ASL void*)(ldst)),        \
                  "v"((unsigned long long)(gsrc))                              \
               : "memory")
#endif

#if __has_builtin(__builtin_amdgcn_s_wait_asynccnt)
#define WAIT_ASYNC(n)                                                          \
  do { __builtin_amdgcn_s_wait_asynccnt(n); asm volatile("" ::: "memory"); } while (0)
#else
#define WAIT_ASYNC(n) asm volatile("s_wait_asynccnt %0" :: "n"(n) : "memory")
#endif

#if __has_builtin(__builtin_amdgcn_s_wait_dscnt)
#define WAIT_DS()                                                              \
  do { __builtin_amdgcn_s_wait_dscnt(0); asm volatile("" ::: "memory"); } while (0)
#else
#define WAIT_DS() asm volatile("s_wait_dscnt 0x0" ::: "memory")
#endif

#define NEG_INF (-__builtin_inff())

constexpr int KCAP   = 1024;   // max top-k (matches reference K_CAP)
constexpr int NTHR   = 1024;   // threads per row-workgroup (32 wave32 waves)
constexpr int CHUNK4 = 1024;   // float4 elements per streamed chunk (16 KB)
constexpr int CAND   = 2048;   // candidate buffer (>= K, sorted by bitonic net)

// Order-preserving float <-> uint key (descending value == descending key).
__device__ __forceinline__ unsigned key_of(float f) {
  unsigned u = __float_as_uint(f);
  return (u & 0x80000000u) ? ~u : (u | 0x80000000u);
}
__device__ __forceinline__ float val_of(unsigned u) {
  unsigned b = (u & 0x80000000u) ? (u & 0x7fffffffu) : ~u;
  return __uint_as_float(b);
}

// Stream one row (numV4 float4's) through a double-buffered async->LDS
// pipeline.  Each lane owns a private 16B slot per buffer, so the pipeline
// needs no workgroup barriers: per-wave s_wait_asynccnt is sufficient.
template <typename F>
__device__ __forceinline__ void stream_row(const float* __restrict__ row,
                                           float* stage, int tid, int numV4,
                                           F consume) {
  const int nChunks = (numV4 + CHUNK4 - 1) / CHUNK4;
  auto issue = [&](int c) {
    int idx4 = c * CHUNK4 + tid;
    if (idx4 >= numV4) idx4 = numV4 - 1;  // clamp: safe addr, never consumed
    const float* g = row + (size_t)idx4 * 4;
    float* l = stage + (size_t)(c & 1) * (CHUNK4 * 4) + (size_t)tid * 4;
    ASYNC_COPY_B128(g, l);
  };
  WAIT_DS();  // previous LDS reads of the stage buffers have retired
  issue(0);
  if (nChunks > 1) issue(1);
  for (int c = 0; c < nChunks; ++c) {
    if (c + 1 < nChunks) { WAIT_ASYNC(1); } else { WAIT_ASYNC(0); }
    const int idx4 = c * CHUNK4 + tid;
    if (idx4 < numV4) {
      const float* l = stage + (size_t)(c & 1) * (CHUNK4 * 4) + (size_t)tid * 4;
      const float4 x = *reinterpret_cast<const float4*>(l);
      const int e = idx4 * 4;
      consume(x.x, e + 0);
      consume(x.y, e + 1);
      consume(x.z, e + 2);
      consume(x.w, e + 3);
    }
    if (c + 2 < nChunks) { WAIT_DS(); issue(c + 2); }
  }
}

// Pre-fill the full-vocab canvas with -inf (vectorized grid-stride).
__global__ void fill_neg_inf(float4* __restrict__ p, long long n4) {
  long long i = (long long)blockIdx.x * blockDim.x + threadIdx.x;
  const long long stride = (long long)gridDim.x * blockDim.x;
  float4 v;
  v.x = v.y = v.z = v.w = NEG_INF;
  for (; i < n4; i += stride) p[i] = v;
}

__global__ __launch_bounds__(NTHR, 1) void topk_topp_sample_scatter(
    const float* __restrict__ logits, const int* __restrict__ top_ks,
    const float* __restrict__ top_ps, const float* __restrict__ q,
    const int* __restrict__ need_flag, float* __restrict__ out_idx,
    float* __restrict__ canvas, int V) {

  union Post {
    float stage[2][CHUNK4 * 4];                               // 32 KB stream buffers
    struct { float e[NTHR]; float sc[NTHR]; int ir[NTHR]; } p; // reused afterwards
  };
  __shared__ __align__(16) Post sm;
  __shared__ unsigned hist[256];
  __shared__ unsigned long long cand[CAND];                    // {~key, idx} packed
  __shared__ unsigned s_pfx, s_rem, s_cnt;

  const int tid = threadIdx.x;
  const int b = blockIdx.x;
  const float* row = logits + (size_t)b * V;
  const int numV4 = V >> 2;
  float* stage = &sm.stage[0][0];

  // ---- Phase 1: radix-select the KCAP-th largest key (4 x 8-bit passes) ----
  if (tid == 0) { s_rem = KCAP; s_pfx = 0; }
  __syncthreads();
  unsigned pfx = 0;
  for (int pass = 0; pass < 4; ++pass) {
    if (tid < 256) hist[tid] = 0;
    __syncthreads();
    const int shift = 24 - 8 * pass;
    const unsigned cur = pfx;
    stream_row(row, stage, tid, numV4, [&](float f, int) {
      unsigned u = key_of(f);
      if (pass == 0 || (u >> (shift + 8)) == cur)
        atomicAdd(&hist[(u >> shift) & 0xFFu], 1u);   // ds_add_u32
    });
    __syncthreads();
    if (tid == 0) {
      unsigned rem = s_rem, cum = 0, bin = 0;
      for (int i = 255; i >= 0; --i) {
        unsigned c = hist[i];
        if (cum + c >= rem) { bin = (unsigned)i; break; }
        cum += c;
      }
      s_rem = rem - cum;
      s_pfx = (cur << 8) | bin;
    }
    __syncthreads();
    pfx = s_pfx;
  }
  const unsigned T = pfx;  // exact key of the KCAP-th largest element

  // ---- Phase 2: compact every element with key >= T ----
  cand[tid] = ~0ULL;
  cand[tid + NTHR] = ~0ULL;
  if (tid == 0) s_cnt = 0;
  __syncthreads();
  stream_row(row, stage, tid, numV4, [&](float f, int e) {
    unsigned u = key_of(f);
    if (u >= T) {
      unsigned slot = atomicAdd(&s_cnt, 1u);
      if (slot < CAND)
        cand[slot] = ((unsigned long long)(~u) << 32) | (unsigned)e;
    }
  });
  __syncthreads();

  // ---- Phase 3: bitonic sort 2048 {~key,idx} ascending == stable top-k ----
  for (unsigned k = 2; k <= CAND; k <<= 1) {
    for (unsigned j = k >> 1; j >= 1; j >>= 1) {
      const unsigned i = ((tid & ~(j - 1)) << 1) | (tid & (j - 1));
      const unsigned pr = i | j;
      const unsigned long long a = cand[i], c = cand[pr];
      const bool up = ((i & k) == 0);
      if ((a > c) == up) { cand[i] = c; cand[pr] = a; }
      __syncthreads();
    }
  }

  // ---- Phase 4: softmax over top-k prefix, top-p cut, ratio-argmax sample ----
  const unsigned long long mine = cand[tid];
  const float myval = val_of(~(unsigned)(mine >> 32));

  int krow = top_ks[b];
  if (krow < 1) krow = 1;
  if (krow > KCAP) krow = KCAP;
  const float p = top_ps[b];
  const bool usep = (p > 0.0f) && (p < 1.0f);

  const float vmax = val_of(~(unsigned)(cand[0] >> 32));  // row max (sorted)
  const float ev = (tid < krow) ? __expf(myval - vmax) : 0.0f;
  __syncthreads();                 // stage buffers fully retired; reuse as Post
  sm.p.e[tid] = ev;
  sm.p.sc[tid] = ev;
  __syncthreads();
  // Hillis-Steele inclusive scan (1024)
  for (int off = 1; off < NTHR; off <<= 1) {
    const float v = sm.p.sc[tid];
    const float add = (tid >= off) ? sm.p.sc[tid - off] : 0.0f;
    __syncthreads();
    sm.p.sc[tid] = v + add;
    __syncthreads();
  }
  const float S = sm.p.sc[krow - 1];           // softmax denom over top-k
  const float cum = sm.p.sc[tid] / S;          // cumulative prob
  const bool hit = (tid < krow) && (cum > p);
  sm.p.ir[tid] = hit ? tid : 0x7fffffff;
  __syncthreads();
  for (int s = NTHR / 2; s > 0; s >>= 1) {     // min-index reduce
    if (tid < s) {
      const int o = sm.p.ir[tid + s];
      if (o < sm.p.ir[tid]) sm.p.ir[tid] = o;
    }
    __syncthreads();
  }
  const int first = sm.p.ir[0];
  const int n = usep ? ((first != 0x7fffffff) ? first + 1 : krow) : krow;
  const float S2 = sm.p.sc[n - 1];             // renorm denom over kept prefix
  const float kp = (tid < n) ? sm.p.e[tid] / S2 : 0.0f;
  const float qv = fabsf(q[(size_t)b * V + tid]) + 1e-8f;  // q by POSITION
  const float score = (tid < n) ? (kp / qv) : NEG_INF;
  __syncthreads();
  sm.p.e[tid] = score;
  sm.p.ir[tid] = tid;
  __syncthreads();
  for (int s = NTHR / 2; s > 0; s >>= 1) {     // argmax, lowest index on tie
    if (tid < s) {
      const float so = sm.p.e[tid + s]; const int io = sm.p.ir[tid + s];
      const float sc = sm.p.e[tid];     const int ic = sm.p.ir[tid];
      if (so > sc || (so == sc && io < ic)) { sm.p.e[tid] = so; sm.p.ir[tid] = io; }
    }
    __syncthreads();
  }
  const int sample = sm.p.ir[0];
  if (tid == 0)
    out_idx[b] = (float)(unsigned)(cand[sample] & 0xffffffffu);

  // ---- Phase 5: scatter kept logits into the -inf canvas ----
  if (need_flag[0] != 0) {
    float* orow = canvas + (size_t)b * V;
    for (int i = tid; i < n; i += NTHR) {
      const unsigned long long c = cand[i];
      orow[(unsigned)(c & 0xffffffffu)] = val_of(~(unsigned)(c >> 32));
    }
  }
}

extern "C" void kernel_launch(void* const* d_in, const int* in_sizes, int n_in,
                              void* d_out, int out_size, void* d_ws, size_t ws_size,
                              hipStream_t stream) {
  (void)n_in; (void)out_size; (void)d_ws; (void)ws_size;
  const float* logits = (const float*)d_in[0];
  const int*   top_ks = (const int*)d_in[1];
  const float* top_ps = (const float*)d_in[2];
  const float* q      = (const float*)d_in[3];
  const int*   need   = (const int*)d_in[4];

  const int B = in_sizes[1];
  const int V = in_sizes[0] / B;

  float* out_idx = (float*)d_out;      // [B] selected vocab indices
  float* canvas  = out_idx + B;        // [B, V] kept-logit canvas

  const long long n4 = ((long long)B * (long long)V) >> 2;
  fill_neg_inf<<<2048, 256, 0, stream>>>((float4*)canvas, n4);
  topk_topp_sample_scatter<<<B, NTHR, 0, stream>>>(logits, top_ks, top_ps, q,
                                                   need, out_idx, canvas, V);
}